// RTRLRTU_41274635714700
// MI455X (gfx1250) — compile-verified
//
#include <hip/hip_runtime.h>
#include <hip/hip_bf16.h>

typedef __attribute__((ext_vector_type(2))) float v2f;
typedef __attribute__((ext_vector_type(8))) float v8f;

#define HH 256   // hidden size
#define DD 256   // input vocab size
#define TT 512   // sequence length

// -------------------------------------------------------------------------
// Kernel 1: Bx[which][t][h] = sum_d x[t][d] * Bc[which][h][d]
// One wave32 per 16x16 (t,h) output tile, V_WMMA_F32_16X16X4_F32 over K=D.
// grid = (T/16, H/16, 2), block = 32 (EXEC all-1s as WMMA requires).
// -------------------------------------------------------------------------
__global__ void rtu_proj_wmma(const float* __restrict__ x,
                              const float* __restrict__ Bc1,
                              const float* __restrict__ Bc2,
                              float* __restrict__ bx)   // [2][T][H]
{
    const int lane   = threadIdx.x;      // 0..31
    const int mt     = blockIdx.x;       // t-tile
    const int nt     = blockIdx.y;       // h-tile
    const int which  = blockIdx.z;       // 0 -> B_c1, 1 -> B_c2

    const float* __restrict__ B   = which ? Bc2 : Bc1;
    float*       __restrict__ out = bx + which * (TT * HH);

    const int laneLo = lane & 15;
    const int laneHi = lane >> 4;        // 0 or 1

    // ISA layout, 32-bit A 16x4: lanes 0-15 hold K={0,1}, lanes 16-31 K={2,3};
    // B 4x16 mirrors with N striped across lanes.
    const int arow = mt * 16 + laneLo;                 // t row of X
    const int bcol = nt * 16 + laneLo;                 // h row of Bc (column of B-matrix)

    const float* __restrict__ aptr = x + arow * DD + 2 * laneHi;
    const float* __restrict__ bptr = B + bcol * DD + 2 * laneHi;

    v8f acc = {};
    #pragma unroll 8
    for (int k = 0; k < DD; k += 4) {
        v2f a = *(const v2f*)(aptr + k);
        v2f b = *(const v2f*)(bptr + k);
        // 8 args: (neg_a, A, neg_b, B, c_mod, C, reuse_a, reuse_b)
        acc = __builtin_amdgcn_wmma_f32_16x16x4_f32(
            false, a, false, b, (short)0, acc, false, false);
    }

    // C/D 16x16 f32 layout: lane group selects M offset (0 or 8), N = lane&15.
    #pragma unroll
    for (int r = 0; r < 8; ++r) {
        const int m = r + 8 * laneHi;               // row within tile (t)
        const int tIdx = mt * 16 + m;
        const int hIdx = nt * 16 + laneLo;
        out[tIdx * HH + hIdx] = acc[r];
    }
}

// -------------------------------------------------------------------------
// Kernel 2: sequential scan. Only h1/h2 are live in the reference output
// (JAX DCEs sr*/st*/sB* since the final carry is discarded).
//   h1n = c*h1 - s*h2 + gamma*Bx1[t][h]
//   h2n = c*h2 + s*h1 + gamma*Bx2[t][h]
//   out[t][h] = h1n ; out[t][H+h] = h2n
// One thread per hidden unit; coalesced Bx reads and out writes.
// -------------------------------------------------------------------------
__global__ void rtu_scan(const float* __restrict__ lamda,
                         const float* __restrict__ theta,
                         const float* __restrict__ bx,   // [2][T][H]
                         float* __restrict__ out)        // [T][2H]
{
    const int h = blockIdx.x * blockDim.x + threadIdx.x;
    if (h >= HH) return;

    const float la = lamda[h];
    const float th = theta[h];
    const float el    = expf(la);
    const float y     = expf(-el);
    const float z     = expf(th);
    const float gamma = sqrtf(1.0f - y * y);
    const float c     = y * cosf(z);
    const float s     = y * sinf(z);

    const float* __restrict__ bx1 = bx;
    const float* __restrict__ bx2 = bx + TT * HH;

    float h1 = 0.0f, h2 = 0.0f;
    for (int t = 0; t < TT; ++t) {
        // hint the load stream ahead (lowers to global_prefetch_b8)
        if (t + 16 < TT) {
            __builtin_prefetch(&bx1[(t + 16) * HH + h], 0, 1);
            __builtin_prefetch(&bx2[(t + 16) * HH + h], 0, 1);
        }
        const float b1 = bx1[t * HH + h];
        const float b2 = bx2[t * HH + h];
        const float h1n = fmaf(c, h1, fmaf(-s, h2, gamma * b1));
        const float h2n = fmaf(c, h2, fmaf( s, h1, gamma * b2));
        out[t * (2 * HH) + h]      = h1n;
        out[t * (2 * HH) + HH + h] = h2n;
        h1 = h1n;
        h2 = h2n;
    }
}

// -------------------------------------------------------------------------
// Launch: inputs in setup_inputs() order:
//   0: x_sequence [T,D] f32   1: lamda [H] f32   2: theta [H] f32
//   3: B_c1 [H,D] f32         4: B_c2 [H,D] f32
// d_out: [T, 2H] f32. d_ws: Bx scratch, 2*T*H floats = 1 MB.
// -------------------------------------------------------------------------
extern "C" void kernel_launch(void* const* d_in, const int* in_sizes, int n_in,
                              void* d_out, int out_size, void* d_ws, size_t ws_size,
                              hipStream_t stream) {
    const float* x   = (const float*)d_in[0];
    const float* lam = (const float*)d_in[1];
    const float* th  = (const float*)d_in[2];
    const float* Bc1 = (const float*)d_in[3];
    const float* Bc2 = (const float*)d_in[4];
    float* out = (float*)d_out;
    float* bx  = (float*)d_ws;   // 2*T*H floats

    dim3 g1(TT / 16, HH / 16, 2);
    rtu_proj_wmma<<<g1, 32, 0, stream>>>(x, Bc1, Bc2, bx);

    rtu_scan<<<(HH + 255) / 256, 256, 0, stream>>>(lam, th, bx, out);
}